// FlexPooling_23708219474789
// MI455X (gfx1250) — compile-verified
//
#include <hip/hip_runtime.h>

// FlexPooling: out[b,d,n] = max_k features[b, d, nbr[b,k,n]]
// B=4, D=128, N=32768, K=16  (float32 features/output, int32 indices)

constexpr int Bc = 4;
constexpr int Dc = 128;
constexpr int Nc = 32768;
constexpr int Kc = 16;

constexpr int TILE   = 256;  // n-values per block (== blockDim.x, 8 wave32s)
constexpr int DCHUNK = 16;   // d-values per block

typedef __attribute__((address_space(3))) int lds_i32;

__global__ __launch_bounds__(TILE) void flexpool_gather_max(
    const float* __restrict__ feats,
    const int*   __restrict__ nbr,
    float*       __restrict__ out)
{
    const int tid = threadIdx.x;
    const int n0  = blockIdx.x * TILE;    // n-tile
    const int d0  = blockIdx.y * DCHUNK;  // d-chunk
    const int b   = blockIdx.z;           // batch

    __shared__ int sidx[Kc * TILE];

    // ---- Stage the [K x TILE] neighborhood tile into LDS via CDNA5 async DMA.
    // GVS addressing: saddr = uniform 64-bit base, vaddr = per-lane byte offset.
    // Each lane DMAs exactly the K indices it will consume below, so a single
    // s_wait_asynccnt 0 (no workgroup barrier) makes the data safe to read.
    const int* gbase = nbr + (size_t)b * Kc * Nc;                  // uniform
    const unsigned ldsbase = (unsigned)(size_t)(lds_i32*)&sidx[0]; // LDS byte offset
    #pragma unroll
    for (int k = 0; k < Kc; ++k) {
        unsigned voff = (unsigned)(((k * Nc) + n0 + tid) * 4);     // <= 2 MB
        unsigned loff = ldsbase + (unsigned)((k * TILE + tid) * 4);
        asm volatile("global_load_async_to_lds_b32 %0, %1, %2"
                     :: "v"(loff), "v"(voff), "s"(gbase)
                     : "memory");
    }
    asm volatile("s_wait_asynccnt 0" ::: "memory");

    // Hoist this thread's 16 indices into registers: reused across all DCHUNK
    // d-iterations (and the async DMA freed the vector pipe during staging).
    int idx[Kc];
    #pragma unroll
    for (int k = 0; k < Kc; ++k) idx[k] = sidx[k * TILE + tid];

    // ---- Gather + max. Row base (feats + (b*D+d)*N) is uniform per block ->
    // compiler emits global_load_b32 with SGPR base + scaled VGPR index (GVS),
    // 16 independent L2-hit gathers in flight per d. Output is streamed with
    // non-temporal stores so the 64 MB result does not evict L2-resident
    // features (entire features tensor = 64 MB < 192 MB L2).
    const float* frow = feats + (size_t)(b * Dc + d0) * Nc;
    float*       orow = out   + (size_t)(b * Dc + d0) * Nc + n0 + tid;

    #pragma unroll 2
    for (int d = 0; d < DCHUNK; ++d) {
        const float* f = frow + (size_t)d * Nc;
        float m = f[idx[0]];
        #pragma unroll
        for (int k = 1; k < Kc; ++k) m = fmaxf(m, f[idx[k]]);
        __builtin_nontemporal_store(m, orow + (size_t)d * Nc);
    }
}

extern "C" void kernel_launch(void* const* d_in, const int* in_sizes, int n_in,
                              void* d_out, int out_size, void* d_ws, size_t ws_size,
                              hipStream_t stream) {
    const float* feats = (const float*)d_in[0];   // [B, D, N] float32
    const int*   nbr   = (const int*)d_in[1];     // [B, K, N] int32
    float*       out   = (float*)d_out;           // [B, D, N] float32

    dim3 grid(Nc / TILE, Dc / DCHUNK, Bc);        // 128 x 8 x 4 = 4096 blocks
    flexpool_gather_max<<<grid, dim3(TILE), 0, stream>>>(feats, nbr, out);
}